// FDTDKernel2D_32040456028766
// MI455X (gfx1250) — compile-verified
//
#include <hip/hip_runtime.h>
#include <hip/hip_bf16.h>

// FDTD 2D single-pass kernel for MI455X (gfx1250).
// Memory-bound: ~0.8 GB traffic -> ~35us at 23.3 TB/s. f32 throughout (must match
// reference numerics). CDNA5 path used: async global->LDS staging (ASYNCcnt) for
// the halo'd stencil tiles, computed out of LDS, non-temporal output stores.

#define NXD 4096
#define NYD 4096
#define NBD 2
#define TPB 256

constexpr double DT_   = 1e-11;
constexpr double DX_   = 0.01;
constexpr double EPS0_ = 8.854e-12;
constexpr double MU0_  = 4.0 * 3.1415926 * 1e-7;
constexpr float  C_EPS_F = (float)(DT_ / EPS0_ / DX_);
constexpr float  C_MU_F  = (float)(DT_ / MU0_ / DX_);

#if __has_builtin(__builtin_amdgcn_global_load_async_to_lds_b32)
#define HAS_ASYNC_LDS 1
#else
#define HAS_ASYNC_LDS 0
#endif

typedef __attribute__((address_space(1))) int g_int;
typedef __attribute__((address_space(3))) int l_int;

__device__ __forceinline__ void stage1(const float* __restrict__ g, float* l) {
#if HAS_ASYNC_LDS
  __builtin_amdgcn_global_load_async_to_lds_b32(
      (g_int*)g, (l_int*)l, /*imm offset=*/0, /*cpol=*/0);
#else
  *l = *g;
#endif
}

__device__ __forceinline__ void wait_stage() {
#if HAS_ASYNC_LDS
#if __has_builtin(__builtin_amdgcn_s_wait_asynccnt)
  __builtin_amdgcn_s_wait_asynccnt(0);
#else
  asm volatile("s_wait_asynccnt 0" ::: "memory");
#endif
#endif
  __syncthreads();
}

__global__ __launch_bounds__(TPB) void fdtd2d_kernel(
    const float* __restrict__ Ez, const float* __restrict__ Hx,
    const float* __restrict__ Hy,
    float* __restrict__ oEz, float* __restrict__ oHx, float* __restrict__ oHy)
{
  // LDS staging: Ez rows i-1,i,i+1 over columns [j0-1, j0+256] (258 each),
  // Hx row i over [j0-1, j0+255] (257), Hy rows i-1,i over [j0, j0+255].
  __shared__ float s_ezm[258];
  __shared__ float s_ezc[258];
  __shared__ float s_ezp[258];
  __shared__ float s_hx[257];
  __shared__ float s_hym[TPB];
  __shared__ float s_hyc[TPB];

  const int t  = threadIdx.x;
  const int j0 = blockIdx.x * TPB;   // column-tile origin
  const int i  = blockIdx.y;         // row
  const int b  = blockIdx.z;         // batch
  const int j  = j0 + t;

  // Clamped neighbor rows: at a clamped edge the finite difference is exactly 0,
  // reproducing the reference's "boundary slice untouched" semantics.
  const int im = (i > 0)        ? (i - 1) : 0;
  const int ip = (i < NXD - 1)  ? (i + 1) : (NXD - 1);

  const size_t plane = (size_t)b * NXD * NYD;
  const float* ezm_row = Ez + plane + (size_t)im * NYD;
  const float* ezc_row = Ez + plane + (size_t)i  * NYD;
  const float* ezp_row = Ez + plane + (size_t)ip * NYD;
  const float* hxc_row = Hx + plane + (size_t)i  * NYD;
  const float* hym_row = Hy + plane + (size_t)im * NYD;
  const float* hyc_row = Hy + plane + (size_t)i  * NYD;

  // ---- stage tiles (async global -> LDS) ----
  for (int l = t; l < 258; l += TPB) {
    int col = j0 - 1 + l;
    col = col < 0 ? 0 : (col > NYD - 1 ? NYD - 1 : col);  // clamp both ends
    stage1(ezm_row + col, &s_ezm[l]);
    stage1(ezc_row + col, &s_ezc[l]);
    stage1(ezp_row + col, &s_ezp[l]);
  }
  for (int l = t; l < 257; l += TPB) {
    int col = j0 - 1 + l;              // max = j0+255 <= NYD-1, only left clamp needed
    col = col < 0 ? 0 : col;
    stage1(hxc_row + col, &s_hx[l]);
  }
  stage1(hym_row + j, &s_hym[t]);
  stage1(hyc_row + j, &s_hyc[t]);

  wait_stage();   // s_wait_asynccnt 0 + workgroup barrier

  // ---- compute (all operands from LDS; stride-1, conflict-free) ----
  const int l = t + 1;                 // position of column j in the halo'd rows
  const float ez_c    = s_ezc[l];
  const float ez_jm   = s_ezc[l - 1];
  const float ez_jp   = s_ezc[l + 1];
  const float ez_im   = s_ezm[l];
  const float ez_imjp = s_ezm[l + 1];
  const float ez_ip   = s_ezp[l];
  const float ez_ipjm = s_ezp[l - 1];
  const float hx_c    = s_hx[l];
  const float hx_jm   = s_hx[l - 1];
  const float hy_c    = s_hyc[t];
  const float hy_im   = s_hym[t];

  // H updates (clamped diffs vanish at i=NX-1 / j=NY-1 -> H unchanged there)
  const float hx_new    = hx_c  - C_MU_F * (ez_ip   - ez_c);
  const float hy_new    = hy_c  + C_MU_F * (ez_jp   - ez_c);
  const float hx_jm_new = hx_jm - C_MU_F * (ez_ipjm - ez_jm);
  const float hy_im_new = hy_im + C_MU_F * (ez_imjp - ez_im);

  // Ez update uses NEW H fields; only interior (i>=1, j>=1) is updated.
  const float curl   = (hy_new - hy_im_new) - (hx_new - hx_jm_new);
  const float ez_new = (i > 0 && j > 0) ? (ez_c + C_EPS_F * curl) : ez_c;

  const size_t idx = plane + (size_t)i * NYD + j;
  __builtin_nontemporal_store(ez_new, oEz + idx);
  __builtin_nontemporal_store(hx_new, oHx + idx);
  __builtin_nontemporal_store(hy_new, oHy + idx);
}

extern "C" void kernel_launch(void* const* d_in, const int* in_sizes, int n_in,
                              void* d_out, int out_size, void* d_ws, size_t ws_size,
                              hipStream_t stream) {
  (void)in_sizes; (void)n_in; (void)out_size; (void)d_ws; (void)ws_size;
  const float* Ez = (const float*)d_in[0];
  const float* Hx = (const float*)d_in[1];
  const float* Hy = (const float*)d_in[2];

  const size_t N = (size_t)NBD * NXD * NYD;
  float* oEz = (float*)d_out;        // outputs concatenated in return order
  float* oHx = oEz + N;
  float* oHy = oHx + N;

  dim3 grid(NYD / TPB, NXD, NBD);    // (16, 4096, 2)
  dim3 block(TPB);
  fdtd2d_kernel<<<grid, block, 0, stream>>>(Ez, Hx, Hy, oEz, oHx, oHy);
}